// KeypointOnPCLoss_66022237274634
// MI455X (gfx1250) — compile-verified
//
#include <hip/hip_runtime.h>
#include <math.h>

// KeypointOnPCLoss for MI455X (gfx1250).
// NN search via V_WMMA_F32_16X16X4_F32 with the pc^2 term folded into the
// spare K=3 slot:
//   A (16x4) = [-2kx, -2ky, -2kz, 1]     (per keypoint row)
//   B (4x16) = [ x,    y,    z,  pc^2]   (per target point column)
//   C        = inline 0
//   D        = pc^2 - 2*cross = d2 - kp^2   (argmin-equivalent to d2)
// Each 256-thread block owns one (batch, 16-keypoint) tile; its 8 waves split
// N=16384 into 2048-point chunks (128 WMMA tiles each). Argmin reductions use
// order-preserving (value,index) u64 keys so every merge is a branchless min.

typedef float v2f __attribute__((ext_vector_type(2)));
typedef float v8f __attribute__((ext_vector_type(8)));

namespace {
constexpr int kB = 4;
constexpr int kM = 1024;
constexpr int kN = 16384;
constexpr float kEPS = 1e-7f;
constexpr int kWavesPerBlock = 8;           // 256 threads
constexpr int kChunk = kN / kWavesPerBlock; // 2048 points per wave
}

__global__ void zero_out_kernel(float* out) { out[0] = 0.0f; }

// Order-preserving f32 -> u32 map: a < b  <=>  key(a) < key(b) (no NaNs here).
__device__ __forceinline__ unsigned int f32_order_key(float f) {
  unsigned int u = __float_as_uint(f);
  return u ^ (((int)u >> 31) | 0x80000000u);
}

__global__ __launch_bounds__(256) void keypoint_pc_loss_kernel(
    const float* __restrict__ kp,   // B,3,M
    const float* __restrict__ pc,   // B,3,N
    const float* __restrict__ sn,   // B,3,N
    float* __restrict__ out)        // scalar
{
  __shared__ unsigned long long sKey[kWavesPerBlock * 16];

  const int tid  = threadIdx.x;
  const int wave = tid >> 5;
  const int lane = tid & 31;
  const int half = lane >> 4;   // A/B K-halves; D rows v (half 0) / v+8 (half 1)
  const int l    = lane & 15;   // tile column (and keypoint row for A)

  const int b  = blockIdx.x >> 6;          // 64 M-tiles per batch
  const int m0 = (blockIdx.x & 63) << 4;   // first keypoint row of this tile

  const float* kpb = kp + (size_t)b * 3 * kM;
  const float* pcb = pc + (size_t)b * 3 * kN;
  const float* snb = sn + (size_t)b * 3 * kN;

  // A tile (16x4 f32): lanes 0-15 hold K=0,1; lanes 16-31 hold K=2,3.
  v2f a;
  if (half == 0) {
    a.x = -2.0f * kpb[0 * kM + m0 + l];
    a.y = -2.0f * kpb[1 * kM + m0 + l];
  } else {
    a.x = -2.0f * kpb[2 * kM + m0 + l];
    a.y = 1.0f;                      // multiplies B's pc^2 row
  }

  float mn[8];
  int   ix[8];
#pragma unroll
  for (int v = 0; v < 8; ++v) { mn[v] = 3.4e38f; ix[v] = 0; }

  const int nBeg = wave * kChunk;
  const int nEnd = nBeg + kChunk;

#pragma unroll 2
  for (int n0 = nBeg; n0 < nEnd; n0 += 16) {
    const int n = n0 + l;
    const float x = pcb[0 * kN + n];
    const float y = pcb[1 * kN + n];
    const float z = pcb[2 * kN + n];
    const float pc2 = x * x + y * y + z * z;

    // B tile (4x16 f32): lanes 0-15: K=0,1 = x,y; lanes 16-31: K=2,3 = z,pc^2.
    v2f bt;
    bt.x = half ? z : x;
    bt.y = half ? pc2 : y;

    v8f c = {};  // inline 0 accumulator
    v8f d = __builtin_amdgcn_wmma_f32_16x16x4_f32(
        false, a, false, bt, (short)0, c, false, false);

#pragma unroll
    for (int v = 0; v < 8; ++v) {
      const float dv = d[v];
      if (dv < mn[v]) { mn[v] = dv; ix[v] = n; }  // strict '<': first index wins
    }
  }

  // Pack (value, index) into order-preserving u64 keys: min-key == argmin with
  // smaller-index tie-break. All further reduction is branchless u64 min.
  unsigned long long key[8];
#pragma unroll
  for (int v = 0; v < 8; ++v)
    key[v] = ((unsigned long long)f32_order_key(mn[v]) << 32) |
             (unsigned int)ix[v];

  // Butterfly across the 16 lanes of each half (xor masks < 16 stay in-half).
#pragma unroll
  for (int off = 1; off < 16; off <<= 1) {
#pragma unroll
    for (int v = 0; v < 8; ++v) {
      const unsigned long long ok =
          __shfl_xor((unsigned long long)key[v], off, 32);
      key[v] = ok < key[v] ? ok : key[v];
    }
  }

  // Publish this wave's 16 per-row keys (row = half*8 + v).
  if (l == 0) {
#pragma unroll
    for (int v = 0; v < 8; ++v) sKey[wave * 16 + half * 8 + v] = key[v];
  }
  __syncthreads();

  // Wave 0 merges the 8 per-wave partials and runs the epilogue.
  float contrib = 0.0f;
  if (tid < 16) {
    unsigned long long best = sKey[tid];
#pragma unroll
    for (int w = 1; w < kWavesPerBlock; ++w) {
      const unsigned long long ok = sKey[w * 16 + tid];
      best = ok < best ? ok : best;
    }
    const int j = (int)(unsigned int)(best & 0xffffffffull);
    const int m = m0 + tid;
    const float kx = kpb[0 * kM + m], ky = kpb[1 * kM + m], kz = kpb[2 * kM + m];
    const float px = pcb[0 * kN + j], py = pcb[1 * kN + j], pz = pcb[2 * kN + j];
    const float nx = snb[0 * kN + j], ny = snb[1 * kN + j], nz = snb[2 * kN + j];
    const float dx = kx - px, dy = ky - py, dz = kz - pz;
    const float dn  = sqrtf(dx * dx + dy * dy + dz * dz);
    const float inv = 1.0f / (dn + kEPS);
    const float dot = (nx * dx + ny * dy + nz * dz) * inv;
    contrib = dot * dot * (1.0f / (float)(kB * kM));
  }

  if (wave == 0) {
#pragma unroll
    for (int off = 16; off > 0; off >>= 1)
      contrib += __shfl_xor(contrib, off, 32);
    if (lane == 0) atomicAdd(out, contrib);
  }
}

extern "C" void kernel_launch(void* const* d_in, const int* in_sizes, int n_in,
                              void* d_out, int out_size, void* d_ws, size_t ws_size,
                              hipStream_t stream) {
  (void)in_sizes; (void)n_in; (void)d_ws; (void)ws_size; (void)out_size;
  const float* kp = (const float*)d_in[0];
  const float* pc = (const float*)d_in[1];
  const float* sn = (const float*)d_in[2];
  float* out = (float*)d_out;

  zero_out_kernel<<<1, 1, 0, stream>>>(out);
  // B * (M/16) = 4 * 64 = 256 blocks, 8 waves each -> 2048 waves.
  keypoint_pc_loss_kernel<<<kB * (kM / 16), 256, 0, stream>>>(kp, pc, sn, out);
}